// Model3_64630667870272
// MI455X (gfx1250) — compile-verified
//
#include <hip/hip_runtime.h>

typedef __attribute__((ext_vector_type(2))) float v2f;
typedef __attribute__((ext_vector_type(8))) float v8f;

#define D_NODE 5
#define D_EDGE 4
#define D_MSG  10
#define K_EDGE 14   // 2*D_NODE + D_EDGE
#define XSTRIDE 12  // padded msg-accumulator row stride (cols 10,11 stay zero)

// ---------------- zero init (harness poisons d_ws) ----------------
__global__ void zero_kernel(float* __restrict__ p, long n) {
    long i = (long)blockIdx.x * blockDim.x + threadIdx.x;
    if (i < n) p[i] = 0.f;
}

// ---------------- weight prep: zero-pad + transpose into workspace ----------
// wpadT [16 n][16 k], w1padT [16 n][12 k], w2padT [16 n][12 k]
// -> every B-fragment K-pair becomes ONE unconditional global_load_b64.
__global__ void prep_kernel(const float* __restrict__ w_mpl,  // [14,10]
                            const float* __restrict__ w1,     // [10,10]
                            const float* __restrict__ w2,     // [10,5]
                            float* __restrict__ wpadT,
                            float* __restrict__ w1padT,
                            float* __restrict__ w2padT) {
    const int t = threadIdx.x;
    {   // 16x16
        const int n = t >> 4, k = t & 15;
        wpadT[n * 16 + k] = (k < K_EDGE && n < D_MSG) ? w_mpl[k * D_MSG + n] : 0.f;
    }
    if (t < 192) {  // 16x12
        const int n = t / 12, k = t % 12;
        w1padT[n * 12 + k] = (k < 10 && n < 10) ? w1[k * 10 + n] : 0.f;
        w2padT[n * 12 + k] = (k < 10 && n < 5)  ? w2[k * 5 + n]  : 0.f;
    }
}

// ---------------- edge message MLP + scatter-sum (FULL tiles only) ----------
// One wave = one complete 16-edge tile (launcher guarantees coverage and
// n_edges%4==0 for the int4 index loads). Features staged in a zero-padded
// 16x16 LDS tile; A pairs via ds_load_b64, B pairs via global_load_b64,
// 4 chained V_WMMA_F32_16X16X4_F32, then 8 unguarded f32 atomics per lane.
__global__ void edge_kernel(const int* __restrict__ edge_index,   // [2,E]
                            const float* __restrict__ node_attr,  // [N,5]
                            const float* __restrict__ edge_attr,  // [E,4]
                            const float* __restrict__ wpadT,      // [16][16]
                            const float* __restrict__ b_mpl,      // [10]
                            float* __restrict__ x_accum,          // [N,12]
                            long n_edges) {
    __shared__ float atile[8][16][16];   // 8 KB
    const int  lane  = threadIdx.x & 31;
    const int  wave  = threadIdx.x >> 5;
    const int  half  = lane >> 4;
    const int  m     = lane & 15;
    const long ebase = ((long)blockIdx.x * 8 + wave) * 16;
    const int  eli   = (int)(ebase + m);
    const int* __restrict__ dst_idx = edge_index + n_edges;

    // Gather: lane (m,half) fills row m, cols [8*half, 8*half+8).
    float4 lo, hi;
    if (half == 0) {
        const int s = edge_index[eli];
        const int d = dst_idx[eli];
        const float* ps = node_attr + s * D_NODE;
        const float* pd = node_attr + d * D_NODE;
        lo = make_float4(ps[0], ps[1], ps[2], ps[3]);
        hi = make_float4(ps[4], pd[0], pd[1], pd[2]);
    } else {
        const int d = dst_idx[eli];
        const float* pd = node_attr + d * D_NODE;
        const float4 ea = ((const float4*)edge_attr)[eli];  // 16B-aligned row
        lo = make_float4(pd[3], pd[4], ea.x, ea.y);
        hi = make_float4(ea.z, ea.w, 0.f, 0.f);             // pad cols 14,15
    }
    float4* dstp = (float4*)&atile[wave][m][8 * half];
    dstp[0] = lo;
    dstp[1] = hi;
    __syncthreads();

    // ISA f32 16x16x4 fragment layout: lane (m,half), VGPR r -> K = 4c+2*half+r.
    v8f acc = {};
#pragma unroll
    for (int c = 0; c < 4; ++c) {
        const int k0 = 4 * c + 2 * half;
        const v2f a = *(const v2f*)&atile[wave][m][k0];      // ds_load_b64
        const v2f b = *(const v2f*)&wpadT[m * 16 + k0];      // global_load_b64
        acc = __builtin_amdgcn_wmma_f32_16x16x4_f32(false, a, false, b,
                                                    (short)0, acc, false, false);
    }

    // D layout: lane holds column n=m for rows v + 8*half.
    if (m < D_MSG) {
        const float bias = b_mpl[m];
        const int4* dp = (const int4*)(dst_idx + ebase + 8 * half);
        const int4 dA = dp[0], dB = dp[1];
        const int dsts[8] = {dA.x, dA.y, dA.z, dA.w, dB.x, dB.y, dB.z, dB.w};
#pragma unroll
        for (int v = 0; v < 8; ++v) {
            float val = acc[v] + bias;
            val = val > 0.f ? val : 0.f;                     // relu
            atomicAdd(&x_accum[dsts[v] * XSTRIDE + m], val);
        }
    }
}

// ---------------- scalar tail for edges [e0, n_edges) -----------------------
__global__ void edge_tail_kernel(const int* __restrict__ edge_index,
                                 const float* __restrict__ node_attr,
                                 const float* __restrict__ edge_attr,
                                 const float* __restrict__ w_mpl,  // [14,10]
                                 const float* __restrict__ b_mpl,
                                 float* __restrict__ x_accum,      // [N,12]
                                 long e0, long n_edges) {
    const long e = e0 + (long)blockIdx.x * blockDim.x + threadIdx.x;
    if (e >= n_edges) return;
    const int* dst_idx = edge_index + n_edges;
    const int s = edge_index[e];
    const int d = dst_idx[e];
    float f[K_EDGE];
#pragma clang loop unroll(disable)
    for (int i = 0; i < D_NODE; ++i) {
        f[i]          = node_attr[s * D_NODE + i];
        f[D_NODE + i] = node_attr[d * D_NODE + i];
    }
#pragma clang loop unroll(disable)
    for (int i = 0; i < D_EDGE; ++i) f[2 * D_NODE + i] = edge_attr[e * D_EDGE + i];
#pragma clang loop unroll(disable)
    for (int n = 0; n < D_MSG; ++n) {
        float t = b_mpl[n];
#pragma clang loop unroll(disable)
        for (int k = 0; k < K_EDGE; ++k) t += f[k] * w_mpl[k * D_MSG + n];
        t = t > 0.f ? t : 0.f;
        atomicAdd(&x_accum[d * XSTRIDE + n], t);
    }
}

// ---------------- node MLP head + per-graph pooling (FULL tiles only) -------
__global__ void node_kernel(const float* __restrict__ x_accum,  // [N,12]
                            const int* __restrict__ batch,      // [N]
                            const float* __restrict__ w1padT,   // [16][12]
                            const float* __restrict__ b1,       // [10]
                            const float* __restrict__ w2padT,   // [16][12]
                            const float* __restrict__ b2,       // [5]
                            float* __restrict__ g_accum) {      // [G,5]
    __shared__ float htile[8][16][12];   // 6 KB
    const int  lane  = threadIdx.x & 31;
    const int  wave  = threadIdx.x >> 5;
    const int  half  = lane >> 4;
    const int  m     = lane & 15;
    const long nbase = ((long)blockIdx.x * 8 + wave) * 16;
    const int  ri    = (int)(nbase + m);

    v8f acc = {};
#pragma unroll
    for (int c = 0; c < 3; ++c) {
        const int k0 = 4 * c + 2 * half;
        const v2f a = *(const v2f*)&x_accum[ri * XSTRIDE + k0];
        const v2f b = *(const v2f*)&w1padT[m * 12 + k0];
        acc = __builtin_amdgcn_wmma_f32_16x16x4_f32(false, a, false, b,
                                                    (short)0, acc, false, false);
    }

    // bias+relu, transpose through LDS; pad columns 10,11 with zeros.
    if (m < 12) {
        const float bias = (m < 10) ? b1[m] : 0.f;
#pragma unroll
        for (int v = 0; v < 8; ++v) {
            float val = (m < 10) ? (acc[v] + bias) : 0.f;
            htile[wave][v + 8 * half][m] = val > 0.f ? val : 0.f;
        }
    }
    __syncthreads();

    v8f acc2 = {};
#pragma unroll
    for (int c = 0; c < 3; ++c) {
        const int k0 = 4 * c + 2 * half;
        const v2f a = *(const v2f*)&htile[wave][m][k0];      // ds_load_b64
        const v2f b = *(const v2f*)&w2padT[m * 12 + k0];
        acc2 = __builtin_amdgcn_wmma_f32_16x16x4_f32(false, a, false, b,
                                                     (short)0, acc2, false, false);
    }

    if (m < 5) {
        const float bias = b2[m];
        const int4* bp = (const int4*)(batch + nbase + 8 * half);  // 16B aligned
        const int4 bA = bp[0], bB = bp[1];
        const int gs[8] = {bA.x, bA.y, bA.z, bA.w, bB.x, bB.y, bB.z, bB.w};
#pragma unroll
        for (int v = 0; v < 8; ++v) {
            float val = acc2[v] + bias;
            val = val > 0.f ? val : 0.f;
            atomicAdd(&g_accum[gs[v] * 5 + m], val);
        }
    }
}

// ---------------- scalar tail for nodes [r0, n_nodes) -----------------------
__global__ void node_tail_kernel(const float* __restrict__ x_accum,  // [N,12]
                                 const int* __restrict__ batch,
                                 const float* __restrict__ w1, const float* __restrict__ b1,
                                 const float* __restrict__ w2, const float* __restrict__ b2,
                                 float* __restrict__ g_accum,
                                 long r0, long n_nodes) {
    const long r = r0 + (long)blockIdx.x * blockDim.x + threadIdx.x;
    if (r >= n_nodes) return;
    float x[10], h[10];
#pragma clang loop unroll(disable)
    for (int k = 0; k < 10; ++k) x[k] = x_accum[r * XSTRIDE + k];
#pragma clang loop unroll(disable)
    for (int j = 0; j < 10; ++j) {
        float t = b1[j];
#pragma clang loop unroll(disable)
        for (int k = 0; k < 10; ++k) t += x[k] * w1[k * 10 + j];
        h[j] = t > 0.f ? t : 0.f;
    }
    const int g = batch[r];
#pragma clang loop unroll(disable)
    for (int j = 0; j < 5; ++j) {
        float t = b2[j];
#pragma clang loop unroll(disable)
        for (int k = 0; k < 10; ++k) t += h[k] * w2[k * 5 + j];
        t = t > 0.f ? t : 0.f;
        atomicAdd(&g_accum[g * 5 + j], t);
    }
}

// ---------------- per-graph 5->5->1 head (30 FMAs/graph: scalar) ------------
__global__ void graph_kernel(const float* __restrict__ g_accum,  // [G,5]
                             const float* __restrict__ w3, const float* __restrict__ b3,
                             const float* __restrict__ w4, const float* __restrict__ b4,
                             float* __restrict__ out, long n_graphs) {
    const long i = (long)blockIdx.x * blockDim.x + threadIdx.x;
    if (i >= n_graphs) return;
    const int ib = (int)i;
    float h[5];
#pragma unroll
    for (int k = 0; k < 5; ++k) h[k] = g_accum[ib * 5 + k];
    float o = b4[0];
#pragma unroll
    for (int j = 0; j < 5; ++j) {
        float t = b3[j];
#pragma unroll
        for (int k = 0; k < 5; ++k) t += h[k] * w3[k * 5 + j];
        t = t > 0.f ? t : 0.f;
        o += t * w4[j];
    }
    out[ib] = o;
}

extern "C" void kernel_launch(void* const* d_in, const int* in_sizes, int n_in,
                              void* d_out, int out_size, void* d_ws, size_t ws_size,
                              hipStream_t stream) {
    const int*   edge_index = (const int*)  d_in[0];
    const float* node_attr  = (const float*)d_in[1];
    const float* edge_attr  = (const float*)d_in[2];
    const int*   batch      = (const int*)  d_in[3];
    // d_in[4] = num_graphs (device scalar) -- derive G from out_size instead
    const float* w_mpl = (const float*)d_in[5];
    const float* b_mpl = (const float*)d_in[6];
    const float* w1    = (const float*)d_in[7];
    const float* b1    = (const float*)d_in[8];
    const float* w2    = (const float*)d_in[9];
    const float* b2    = (const float*)d_in[10];
    const float* w3    = (const float*)d_in[11];
    const float* b3    = (const float*)d_in[12];
    const float* w4    = (const float*)d_in[13];
    const float* b4    = (const float*)d_in[14];
    float* out = (float*)d_out;

    const long n_edges  = (long)in_sizes[0] / 2;
    const long n_nodes  = (long)in_sizes[1] / D_NODE;
    const long n_graphs = (long)out_size;   // output is [G,1]

    // workspace layout
    float* x_accum = (float*)d_ws;                       // N*12
    float* g_accum = x_accum + n_nodes * XSTRIDE;        // G*5
    float* wpadT   = g_accum + n_graphs * 5;             // 256
    float* w1padT  = wpadT + 256;                        // 192
    float* w2padT  = w1padT + 192;                       // 192

    const long zn = n_nodes * XSTRIDE + n_graphs * 5;
    zero_kernel<<<(int)((zn + 255) / 256), 256, 0, stream>>>(x_accum, zn);
    prep_kernel<<<1, 256, 0, stream>>>(w_mpl, w1, w2, wpadT, w1padT, w2padT);

    // ---- edges: WMMA kernel on complete 128-edge blocks, scalar tail rest ----
    // int4 index loads in the hot kernel need n_edges % 4 == 0.
    const long efull  = ((n_edges & 3) == 0) ? (n_edges / 128) * 128 : 0;
    const long eblk   = efull / 128;
    if (eblk > 0)
        edge_kernel<<<(int)eblk, 256, 0, stream>>>(
            edge_index, node_attr, edge_attr, wpadT, b_mpl, x_accum, n_edges);
    const long etail = n_edges - efull;
    if (etail > 0)
        edge_tail_kernel<<<(int)((etail + 127) / 128), 128, 0, stream>>>(
            edge_index, node_attr, edge_attr, w_mpl, b_mpl, x_accum, efull, n_edges);

    // ---- nodes: WMMA kernel on complete 128-node blocks, scalar tail rest ----
    const long nfull = (n_nodes / 128) * 128;
    const long nblk  = nfull / 128;
    if (nblk > 0)
        node_kernel<<<(int)nblk, 256, 0, stream>>>(
            x_accum, batch, w1padT, b1, w2padT, b2, g_accum);
    const long ntail = n_nodes - nfull;
    if (ntail > 0)
        node_tail_kernel<<<(int)((ntail + 127) / 128), 128, 0, stream>>>(
            x_accum, batch, w1, b1, w2, b2, g_accum, nfull, n_nodes);

    graph_kernel<<<(int)((n_graphs + 255) / 256), 256, 0, stream>>>(
        g_accum, w3, b3, w4, b4, out, n_graphs);
}